// SurvivalGeometryRegularizer_20109036880617
// MI455X (gfx1250) — compile-verified
//
#include <hip/hip_runtime.h>
#include <stdint.h>

#define THREADS 256
#define SPLIT   8        // j-range split -> grid.y
#define JT_MAX  2048     // max j-chunk elements staged in LDS

// CDNA5 async copy: global -> LDS, 16 bytes per lane, tracked by ASYNCcnt.
__device__ __forceinline__ void async_copy_b128(uint32_t lds_off, uint64_t gbase, uint32_t g_off) {
  asm volatile("global_load_async_to_lds_b128 %0, %1, %2"
               :: "v"(lds_off), "v"(g_off), "s"(gbase)
               : "memory");
}

__device__ __forceinline__ void wait_async_zero() {
  asm volatile("s_wait_asynccnt 0" ::: "memory");
}

__global__ void __launch_bounds__(64)
ws_init_kernel(float* tot, unsigned int* cnt) {
  if (threadIdx.x == 0 && blockIdx.x == 0) { *tot = 0.0f; *cnt = 0u; }
}

__global__ void __launch_bounds__(THREADS)
pair_hinge_kernel(const float* __restrict__ risk,
                  const float* __restrict__ timev,
                  const int*   __restrict__ event,
                  float*        __restrict__ ws_tot,
                  unsigned int* __restrict__ ws_cnt,
                  int n) {
  __shared__ __align__(16) float s_risk[JT_MAX];
  __shared__ __align__(16) float s_time[JT_MAX];
  __shared__ float        s_red_t[THREADS / 32];
  __shared__ unsigned int s_red_c[THREADS / 32];

  const int tid = threadIdx.x;
  const int i   = blockIdx.x * THREADS + tid;   // this thread's row
  const int JT  = n / SPLIT;                    // j-chunk length (1024 for B=8192)
  const int j0  = blockIdx.y * JT;

  // ---- stage this block's j-chunk of risk/time into LDS (async tensor path) ----
  {
    const uint32_t lds_r = (uint32_t)(uintptr_t)&s_risk[0]; // low 32 bits = LDS offset
    const uint32_t lds_t = (uint32_t)(uintptr_t)&s_time[0];
    const uint32_t bytes = (uint32_t)JT * 4u;
    const uint32_t gb0   = (uint32_t)j0 * 4u;
    const uint64_t gr    = (uint64_t)(uintptr_t)risk;
    const uint64_t gt    = (uint64_t)(uintptr_t)timev;
    for (uint32_t b = (uint32_t)tid * 16u; b < bytes; b += THREADS * 16u) {
      async_copy_b128(lds_r + b, gr, gb0 + b);
      async_copy_b128(lds_t + b, gt, gb0 + b);
    }
    wait_async_zero();     // each wave waits for its own async writes to land in LDS
    __syncthreads();       // then all waves see the full tile
  }

  // ---- per-row constants; neutralize rows with event != 1 (no divergence) ----
  const float ri = risk[i];
  const float ti = timev[i];
  const int   ev = event[i];
  const float a  = 1.0f + ri;                           // MARGIN + risk_i
  const float tq = (ev == 1) ? ti : __builtin_inff();   // inf => mask always false

  float        total = 0.0f;
  unsigned int cnt   = 0u;

  // ---- 64M-pair core: LDS broadcast reads, float4-unrolled VALU loop ----
  const float4* r4 = (const float4*)s_risk;
  const float4* t4 = (const float4*)s_time;
  const int n4 = JT >> 2;
  for (int q = 0; q < n4; ++q) {
    const float4 rr = r4[q];
    const float4 tt = t4[q];
    if (tq < tt.x) { total += fmaxf(0.0f, a - rr.x); ++cnt; }
    if (tq < tt.y) { total += fmaxf(0.0f, a - rr.y); ++cnt; }
    if (tq < tt.z) { total += fmaxf(0.0f, a - rr.z); ++cnt; }
    if (tq < tt.w) { total += fmaxf(0.0f, a - rr.w); ++cnt; }
  }

  // ---- wave32 reduction, then cross-wave via LDS, then 2 global atomics ----
  for (int off = 16; off > 0; off >>= 1) {
    total += __shfl_xor(total, off, 32);
    cnt   += __shfl_xor(cnt,   off, 32);
  }
  const int wave = tid >> 5;
  if ((tid & 31) == 0) { s_red_t[wave] = total; s_red_c[wave] = cnt; }
  __syncthreads();
  if (tid == 0) {
    float t = 0.0f; unsigned int c = 0u;
#pragma unroll
    for (int w = 0; w < THREADS / 32; ++w) { t += s_red_t[w]; c += s_red_c[w]; }
    atomicAdd(ws_tot, t);
    atomicAdd(ws_cnt, c);
  }
}

__global__ void __launch_bounds__(64)
finalize_kernel(const float* tot, const unsigned int* cnt, float* out) {
  if (threadIdx.x == 0 && blockIdx.x == 0) {
    const unsigned int c = *cnt;
    out[0] = (c == 0u) ? 0.0f : (*tot) / (float)c;
  }
}

extern "C" void kernel_launch(void* const* d_in, const int* in_sizes, int n_in,
                              void* d_out, int out_size, void* d_ws, size_t ws_size,
                              hipStream_t stream) {
  // setup_inputs order: z (unused), risk, time, event
  const float* risk  = (const float*)d_in[1];
  const float* timev = (const float*)d_in[2];
  const int*   event = (const int*)d_in[3];
  float* out = (float*)d_out;
  const int n = in_sizes[1];   // B = 8192

  float*        ws_tot = (float*)d_ws;
  unsigned int* ws_cnt = (unsigned int*)((char*)d_ws + sizeof(float));

  ws_init_kernel<<<1, 64, 0, stream>>>(ws_tot, ws_cnt);
  dim3 grid(n / THREADS, SPLIT);
  pair_hinge_kernel<<<grid, THREADS, 0, stream>>>(risk, timev, event, ws_tot, ws_cnt, n);
  finalize_kernel<<<1, 64, 0, stream>>>(ws_tot, ws_cnt, out);
}